// ProductSpaceLayer_9019431321774
// MI455X (gfx1250) — compile-verified
//
#include <hip/hip_runtime.h>
#include <math.h>

#define DIMV 128
#define TILE_M 16
#define LDS_STRIDE 132   // 132 % 64 = 4 -> conflict-free A-fragment reads
#define EPSF 1e-12f

typedef __attribute__((ext_vector_type(2))) float v2f;
typedef __attribute__((ext_vector_type(8))) float v8f;

// ---------------------------------------------------------------------------
// Fused row-transform + GEMM:  Y[m, j] = f(X)[m, :] . W[j, :]  (+ bias[j])
// MODE 0: f = identity (Euclidean branch, bias added after aggregation)
// MODE 1: f = poincare_logmap0 row scaling
// MODE 2: f = l2_normalize row scaling
// One block = 16-row M tile; 8 waves, each owns a 16-col N tile.
// WMMA: V_WMMA_F32_16X16X4_F32, 32 K-steps over K=128.
// ---------------------------------------------------------------------------
template <int MODE>
__global__ __launch_bounds__(256) void transform_gemm(
    const float* __restrict__ X, const float* __restrict__ W,
    const float* __restrict__ bias, float* __restrict__ Y, int nRows)
{
    __shared__ float sX[TILE_M * LDS_STRIDE];
    __shared__ float sScale[TILE_M];

    const int tid  = threadIdx.x;
    const int lane = tid & 31;
    const int wave = tid >> 5;
    const int m0   = blockIdx.x * TILE_M;

    // --- preload this wave's 32 B fragments of W into registers ------------
    // B(4x16) layout: lanes 0-15 hold K rows {k0,k0+1}, lanes 16-31 {k0+2,k0+3}
    const int j0   = wave * 16;
    const int jj   = j0 + (lane & 15);
    const int ksel = (lane >> 4) << 1;   // 0 or 2
    float bxk[32], byk[32];
#pragma unroll
    for (int k = 0; k < 32; ++k) {
        float2 w2 = *(const float2*)&W[jj * DIMV + k * 4 + ksel];
        bxk[k] = w2.x; byk[k] = w2.y;
    }

    // --- stage 16x128 X tile into LDS (coalesced float4) -------------------
#pragma unroll
    for (int i = 0; i < 2; ++i) {
        int f4  = tid * 2 + i;           // 0..511 float4 slots
        int row = f4 >> 5;               // 0..15
        int col = (f4 & 31) << 2;        // 0..124
        int gr  = m0 + row; if (gr >= nRows) gr = nRows - 1;
        float4 v = *(const float4*)&X[(size_t)gr * DIMV + col];
        *(float4*)&sX[row * LDS_STRIDE + col] = v;
    }
    __syncthreads();

    // --- per-row manifold transform as a row scale -------------------------
    if (MODE != 0) {
        if (tid < TILE_M) {
            float ss = 0.f;
            for (int c = 0; c < DIMV; ++c) {
                float v = sX[tid * LDS_STRIDE + c];
                ss += v * v;
            }
            float nrm = sqrtf(ss);
            float sc;
            if (MODE == 1) {             // logmap0: atanh(clip(n))/n
                float nsafe = fmaxf(nrm, EPSF);
                float ncl   = fminf(fmaxf(nsafe, EPSF), 1.f - 1e-5f);
                sc = atanhf(ncl) / nsafe;
            } else {                     // l2 normalize
                sc = 1.f / fmaxf(nrm, EPSF);
            }
            sScale[tid] = sc;
        }
        __syncthreads();
#pragma unroll
        for (int i = 0; i < 8; ++i) {
            int idx = tid + i * 256;     // 0..2047 over 16x128
            int row = idx >> 7;
            int col = idx & 127;
            sX[row * LDS_STRIDE + col] *= sScale[row];
        }
        __syncthreads();
    }

    // --- WMMA K loop -------------------------------------------------------
    // A(16x4): m = lane&15, lanes 0-15 hold {k0,k0+1}, lanes 16-31 {k0+2,k0+3}
    v8f acc = {};
    const int am = lane & 15;
#pragma unroll
    for (int k = 0; k < 32; ++k) {
        float2 a2 = *(const float2*)&sX[am * LDS_STRIDE + k * 4 + ksel];
        v2f a; a[0] = a2.x; a[1] = a2.y;
        v2f b; b[0] = bxk[k]; b[1] = byk[k];
        acc = __builtin_amdgcn_wmma_f32_16x16x4_f32(
            false, a, false, b, (short)0, acc, false, false);
    }

    // --- store D: VGPR r -> row m0 + r + (lane>=16 ? 8 : 0), col j0+lane%16
    const int jcol = j0 + (lane & 15);
    const float badd = bias ? bias[jcol] : 0.f;
    const int mhi = (lane >> 4) << 3;
#pragma unroll
    for (int r = 0; r < 8; ++r) {
        int m = m0 + r + mhi;
        if (m < nRows) Y[(size_t)m * DIMV + jcol] = acc[r] + badd;
    }
}

// ---------------------------------------------------------------------------
// Edge scatter: 1 wave = 1 edge. Gather 512B row per branch (float4/lane),
// hardware fp32 atomic-add into the per-destination accumulators (= d_out).
// ---------------------------------------------------------------------------
__global__ __launch_bounds__(256) void edge_scatter(
    const int* __restrict__ src, const int* __restrict__ dst,
    const float* __restrict__ tE, const float* __restrict__ tB,
    const float* __restrict__ tS,
    float* __restrict__ accE, float* __restrict__ accB,
    float* __restrict__ accS, float* __restrict__ deg, int nE)
{
    int e    = blockIdx.x * (blockDim.x >> 5) + (threadIdx.x >> 5);
    int lane = threadIdx.x & 31;
    if (e >= nE) return;
    int s = src[e], d = dst[e];
    size_t so = (size_t)s * DIMV + lane * 4;
    size_t dofs = (size_t)d * DIMV + lane * 4;

    float4 ve = *(const float4*)&tE[so];
    float4 vb = *(const float4*)&tB[so];
    float4 vs = *(const float4*)&tS[so];

    float* pe = &accE[dofs];
    float* pb = &accB[dofs];
    float* ps = &accS[dofs];
    unsafeAtomicAdd(pe + 0, ve.x); unsafeAtomicAdd(pe + 1, ve.y);
    unsafeAtomicAdd(pe + 2, ve.z); unsafeAtomicAdd(pe + 3, ve.w);
    unsafeAtomicAdd(pb + 0, vb.x); unsafeAtomicAdd(pb + 1, vb.y);
    unsafeAtomicAdd(pb + 2, vb.z); unsafeAtomicAdd(pb + 3, vb.w);
    unsafeAtomicAdd(ps + 0, vs.x); unsafeAtomicAdd(ps + 1, vs.y);
    unsafeAtomicAdd(ps + 2, vs.z); unsafeAtomicAdd(ps + 3, vs.w);
    if (lane == 0) unsafeAtomicAdd(&deg[d], 1.0f);
}

// ---------------------------------------------------------------------------
// Finalize: 1 wave = 1 node. mean = where(deg>0, sum/max(deg,1), 0), then
// e: +b_e, leaky(0.2);  b: expmap0 (tanh(n)/n);  s: l2 normalize.
// ---------------------------------------------------------------------------
__device__ __forceinline__ float wave_sum(float v) {
#pragma unroll
    for (int off = 16; off > 0; off >>= 1) v += __shfl_xor(v, off, 32);
    return v;
}

__global__ __launch_bounds__(256) void finalize_nodes(
    float* __restrict__ accE, float* __restrict__ accB,
    float* __restrict__ accS, const float* __restrict__ deg,
    const float* __restrict__ b_e, int nNodes)
{
    int node = blockIdx.x * (blockDim.x >> 5) + (threadIdx.x >> 5);
    int lane = threadIdx.x & 31;
    if (node >= nNodes) return;

    float dv  = deg[node];
    float inv = dv > 0.f ? 1.f / fmaxf(dv, 1.f) : 0.f;
    size_t off = (size_t)node * DIMV + lane * 4;

    // Euclidean: leaky_relu(mean + b_e)
    float4 ev  = *(float4*)&accE[off];
    float4 be4 = *(const float4*)&b_e[lane * 4];
    float eo[4] = { ev.x * inv + be4.x, ev.y * inv + be4.y,
                    ev.z * inv + be4.z, ev.w * inv + be4.w };
#pragma unroll
    for (int c = 0; c < 4; ++c) eo[c] = eo[c] >= 0.f ? eo[c] : 0.2f * eo[c];
    *(float4*)&accE[off] = make_float4(eo[0], eo[1], eo[2], eo[3]);

    // Poincare: expmap0(mean)
    float4 bv = *(float4*)&accB[off];
    bv.x *= inv; bv.y *= inv; bv.z *= inv; bv.w *= inv;
    float ssb = wave_sum(bv.x*bv.x + bv.y*bv.y + bv.z*bv.z + bv.w*bv.w);
    float nb  = fmaxf(sqrtf(ssb), EPSF);
    float scb = tanhf(nb) / nb;
    *(float4*)&accB[off] = make_float4(bv.x*scb, bv.y*scb, bv.z*scb, bv.w*scb);

    // Sphere: l2_normalize(mean)
    float4 sv = *(float4*)&accS[off];
    sv.x *= inv; sv.y *= inv; sv.z *= inv; sv.w *= inv;
    float sss = wave_sum(sv.x*sv.x + sv.y*sv.y + sv.z*sv.z + sv.w*sv.w);
    float scs = 1.f / fmaxf(sqrtf(sss), EPSF);
    *(float4*)&accS[off] = make_float4(sv.x*scs, sv.y*scs, sv.z*scs, sv.w*scs);
}

// ---------------------------------------------------------------------------
extern "C" void kernel_launch(void* const* d_in, const int* in_sizes, int n_in,
                              void* d_out, int out_size, void* d_ws, size_t ws_size,
                              hipStream_t stream)
{
    const float* e_emb = (const float*)d_in[0];
    const float* b_emb = (const float*)d_in[1];
    const float* s_emb = (const float*)d_in[2];
    const float* W_e   = (const float*)d_in[3];
    const float* b_e   = (const float*)d_in[4];
    const float* W_b   = (const float*)d_in[5];
    const float* b_b   = (const float*)d_in[6];
    const float* W_s   = (const float*)d_in[7];
    const float* b_s   = (const float*)d_in[8];
    const int*   src   = (const int*)d_in[9];
    const int*   dst   = (const int*)d_in[10];

    const int n  = in_sizes[0] / DIMV;   // nodes
    const int nE = in_sizes[9];          // edges
    const size_t rowsz = (size_t)n * DIMV;

    float* out   = (float*)d_out;
    float* accE  = out;                  // accumulate straight into d_out
    float* accB  = out + rowsz;
    float* accS  = out + 2 * rowsz;

    float* tE  = (float*)d_ws;           // pre-transformed tables
    float* tB  = tE + rowsz;
    float* tS  = tB + rowsz;
    float* deg = tS + rowsz;

    hipMemsetAsync(d_out, 0, (size_t)out_size * sizeof(float), stream);
    hipMemsetAsync(deg, 0, (size_t)n * sizeof(float), stream);

    const int mtiles = (n + TILE_M - 1) / TILE_M;
    transform_gemm<0><<<mtiles, 256, 0, stream>>>(e_emb, W_e, nullptr, tE, n);
    transform_gemm<1><<<mtiles, 256, 0, stream>>>(b_emb, W_b, b_b,     tB, n);
    transform_gemm<2><<<mtiles, 256, 0, stream>>>(s_emb, W_s, b_s,     tS, n);

    edge_scatter<<<(nE + 7) / 8, 256, 0, stream>>>(
        src, dst, tE, tB, tS, accE, accB, accS, deg, nE);

    finalize_nodes<<<(n + 7) / 8, 256, 0, stream>>>(
        accE, accB, accS, deg, b_e, n);
}